// CCMF_54838142435824
// MI455X (gfx1250) — compile-verified
//
#include <hip/hip_runtime.h>
#include <math.h>

typedef float v2f __attribute__((ext_vector_type(2)));
typedef float v8f __attribute__((ext_vector_type(8)));

#define BT 2048      // B*T
#define NG 256       // 4*E gates
#define E  64
#define U  32
#define C  6

__device__ __forceinline__ float sigm(float x) { return 1.0f / (1.0f + __expf(-x)); }

// ---------------------------------------------------------------------------
// Kernel 1: G[m][n] = (x[m]*umask[m]) @ Wih.T + bih + bhh  via WMMA f32 16x16x4
// grid.x = 512, block = 128 (4 waves, one 16x16 C-tile each). 2048 tiles total.
// ---------------------------------------------------------------------------
__global__ __launch_bounds__(128) void gemm_gates_kernel(
    const float* __restrict__ X, const float* __restrict__ um,
    const float* __restrict__ Wih, const float* __restrict__ bih,
    const float* __restrict__ bhh, float* __restrict__ G, int D)
{
    const int wave = threadIdx.x >> 5;
    const int lane = threadIdx.x & 31;
    const int tile = blockIdx.x * 4 + wave;       // [0,2048)
    const int tN = tile & 15;                     // 16 N-tiles (256/16)
    const int tM = tile >> 4;                     // 128 M-tiles (2048/16)
    const int m0 = tM * 16, n0 = tN * 16;
    const int row = lane & 15;
    const int hi  = lane >> 4;                    // 0: K={0,1}, 1: K={2,3}

    const float* Arow = X   + (size_t)(m0 + row) * D;   // A[m0+row][*]
    const float* Brow = Wih + (size_t)(n0 + row) * D;   // B[k][n] = Wih[n][k]
    const float umv = um[m0 + row];

    v8f acc = {};
#pragma unroll 5
    for (int k0 = 0; k0 < D; k0 += 4) {
        v2f a, b;
        a.x = Arow[k0 + 2 * hi    ] * umv;
        a.y = Arow[k0 + 2 * hi + 1] * umv;
        b.x = Brow[k0 + 2 * hi    ];
        b.y = Brow[k0 + 2 * hi + 1];
        acc = __builtin_amdgcn_wmma_f32_16x16x4_f32(
            /*neg_a=*/false, a, /*neg_b=*/false, b,
            /*c_mod=*/(short)0, acc, /*reuse_a=*/false, /*reuse_b=*/false);
    }

    const int n = n0 + row;
    const float bias = bih[n] + bhh[n];
#pragma unroll
    for (int j = 0; j < 8; ++j) {
        int m = m0 + j + 8 * hi;                  // C VGPR j: M=j (lo lanes) / j+8 (hi)
        G[(size_t)m * NG + n] = acc[j] + bias;
    }
}

// ---------------------------------------------------------------------------
// Kernel 2: LSTM recurrence. grid (4 batch-tiles, 2 modalities), 256 threads.
// Each block owns 8 batches; h,c,gates live in LDS; 64 serial steps.
// ---------------------------------------------------------------------------
__global__ __launch_bounds__(256) void lstm_kernel(
    const float* __restrict__ G0, const float* __restrict__ G1,
    const float* __restrict__ Whh0, const float* __restrict__ Whh1,
    float* __restrict__ H0, float* __restrict__ H1)
{
    const float* G   = blockIdx.y ? G1 : G0;
    const float* Whh = blockIdx.y ? Whh1 : Whh0;
    float*       H   = blockIdx.y ? H1 : H0;
    const int b0 = blockIdx.x * 8;
    const int tid = threadIdx.x;

    __shared__ float sh[8][E];    // recurrent h (tanh-ed)
    __shared__ float scc[8][E];   // cell state
    __shared__ float sg[8][NG];   // gates for this step

    for (int p = tid; p < 8 * E; p += 256) {
        sh[p >> 6][p & 63]  = 0.0f;
        scc[p >> 6][p & 63] = 0.0f;
    }
    __syncthreads();

    for (int t = 0; t < 64; ++t) {
        // gates: thread tid computes gate-dim tid for all 8 batches
        float acc[8] = {0.f, 0.f, 0.f, 0.f, 0.f, 0.f, 0.f, 0.f};
        for (int e = 0; e < E; ++e) {
            float w = Whh[tid * E + e];
#pragma unroll
            for (int b = 0; b < 8; ++b) acc[b] += sh[b][e] * w;
        }
#pragma unroll
        for (int b = 0; b < 8; ++b)
            sg[b][tid] = acc[b] + G[((size_t)(b0 + b) * 64 + t) * NG + tid];
        __syncthreads();

        // update: 8*64 = 512 (b,e) pairs, 2 per thread
#pragma unroll
        for (int q = 0; q < 2; ++q) {
            int p = tid + q * 256;
            int b = p >> 6, e = p & 63;
            float gi = sg[b][e];
            float gf = sg[b][64 + e];
            float gg = sg[b][128 + e];
            float go = sg[b][192 + e];
            float cn = sigm(gf) * scc[b][e] + sigm(gi) * tanhf(gg);
            float hn = tanhf(sigm(go) * tanhf(cn));
            scc[b][e] = cn;
            sh[b][e]  = hn;
            H[((size_t)(b0 + b) * 64 + t) * E + e] = hn;
        }
        __syncthreads();
    }
}

// ---------------------------------------------------------------------------
// Kernel 3: normalize measurement kernel rows (U rows of 2E floats)
// ---------------------------------------------------------------------------
__global__ __launch_bounds__(32) void knorm_kernel(
    const float* __restrict__ K, float* __restrict__ knr, float* __restrict__ kni)
{
    int u = threadIdx.x;            // 32 threads, one per measurement
    float ss = 0.0f;
    for (int i = 0; i < 2 * E; ++i) { float v = K[u * 2 * E + i]; ss += v * v; }
    float nrm = sqrtf(ss);
    nrm = fmaxf(nrm, 1e-12f);
    float inv = 1.0f / nrm;
    for (int e = 0; e < E; ++e) {
        knr[u * E + e] = K[u * 2 * E + 2 * e    ] * inv;
        kni[u * E + e] = K[u * 2 * E + 2 * e + 1] * inv;
    }
}

// ---------------------------------------------------------------------------
// Kernel 4: fused tail per (b,t): normalize h, phase lookup, measurement
// (rank-1 simplification), MLP, log-softmax.  One wave32 per (b,t).
// ---------------------------------------------------------------------------
__global__ __launch_bounds__(32) void tail_kernel(
    const float* __restrict__ H0, const float* __restrict__ H1,
    const float* __restrict__ smask, const float* __restrict__ phase_table,
    const float* __restrict__ knr, const float* __restrict__ kni,
    const float* __restrict__ W1, const float* __restrict__ b1,
    const float* __restrict__ W2, const float* __restrict__ b2,
    float* __restrict__ out)
{
    const int bt = blockIdx.x;      // [0, 2048)
    const int tid = threadIdx.x;    // 32 threads = 1 wave

    __shared__ float hn0[E], hn1[E], cv[E], sv[E], mbuf[2 * U], hid[E], pr[C];
    __shared__ int sidx;

    // normalize h0
    {
        const float* hp = H0 + (size_t)bt * E;
        float a = hp[tid], b = hp[tid + 32];
        float ss = a * a + b * b;
        for (int off = 16; off > 0; off >>= 1) ss += __shfl_xor(ss, off, 32);
        float nrm = fmaxf(sqrtf(ss), 1e-12f);
        hn0[tid] = a / nrm; hn0[tid + 32] = b / nrm;
    }
    // normalize h1
    {
        const float* hp = H1 + (size_t)bt * E;
        float a = hp[tid], b = hp[tid + 32];
        float ss = a * a + b * b;
        for (int off = 16; off > 0; off >>= 1) ss += __shfl_xor(ss, off, 32);
        float nrm = fmaxf(sqrtf(ss), 1e-12f);
        hn1[tid] = a / nrm; hn1[tid + 32] = b / nrm;
    }
    // argmax over speakers (first max, as jnp.argmax)
    if (tid == 0) {
        const float* sm = smask + (size_t)bt * 9;
        int bi = 0; float bv = sm[0];
        for (int i = 1; i < 9; ++i) if (sm[i] > bv) { bv = sm[i]; bi = i; }
        sidx = bi;
    }
    __syncthreads();
    {
        int s = sidx;
        float p1 = phase_table[s * E + tid];
        float p2 = phase_table[s * E + tid + 32];
        cv[tid] = __cosf(p1); cv[tid + 32] = __cosf(p2);
        sv[tid] = __sinf(p1); sv[tid + 32] = __sinf(p2);
    }
    __syncthreads();

    // measurement: m_u = (r.kr - im.ki)^2 + (r.ki + im.kr)^2, thread = u
#pragma unroll
    for (int mod = 0; mod < 2; ++mod) {
        const float* hn = mod ? hn1 : hn0;
        float aa = 0.0f, bb = 0.0f;
        for (int e = 0; e < E; ++e) {
            float r  = cv[e] * hn[e];
            float im = sv[e] * hn[e];
            float kr = knr[tid * E + e];
            float ki = kni[tid * E + e];
            aa += r * kr - im * ki;
            bb += r * ki + im * kr;
        }
        mbuf[mod * U + tid] = aa * aa + bb * bb;
    }
    __syncthreads();

    // hid = relu(m @ W1.T + b1), 2 dims per thread
#pragma unroll
    for (int q = 0; q < 2; ++q) {
        int n = tid + q * 32;
        float acc = b1[n];
        for (int k = 0; k < 2 * U; ++k) acc += mbuf[k] * W1[n * 2 * U + k];
        hid[n] = fmaxf(acc, 0.0f);
    }
    __syncthreads();

    if (tid < C) {
        float acc = b2[tid];
        for (int k = 0; k < E; ++k) acc += hid[k] * W2[tid * E + k];
        pr[tid] = tanhf(acc);
    }
    __syncthreads();

    if (tid == 0) {
        float mx = pr[0];
        for (int c = 1; c < C; ++c) mx = fmaxf(mx, pr[c]);
        float se = 0.0f;
        for (int c = 0; c < C; ++c) se += __expf(pr[c] - mx);
        float lse = mx + __logf(se);
        for (int c = 0; c < C; ++c) out[(size_t)bt * C + c] = pr[c] - lse;
    }
}

// ---------------------------------------------------------------------------
extern "C" void kernel_launch(void* const* d_in, const int* in_sizes, int n_in,
                              void* d_out, int out_size, void* d_ws, size_t ws_size,
                              hipStream_t stream)
{
    (void)in_sizes; (void)n_in; (void)out_size; (void)ws_size;

    const float* x0     = (const float*)d_in[0];
    const float* x1     = (const float*)d_in[1];
    const float* smask  = (const float*)d_in[2];
    const float* umask  = (const float*)d_in[3];
    const float* W_ih0  = (const float*)d_in[4];
    const float* W_hh0  = (const float*)d_in[5];
    const float* b_ih0  = (const float*)d_in[6];
    const float* b_hh0  = (const float*)d_in[7];
    const float* W_ih1  = (const float*)d_in[8];
    const float* W_hh1  = (const float*)d_in[9];
    const float* b_ih1  = (const float*)d_in[10];
    const float* b_hh1  = (const float*)d_in[11];
    const float* ptab   = (const float*)d_in[12];
    const float* mkern  = (const float*)d_in[13];
    const float* W1     = (const float*)d_in[14];
    const float* b1     = (const float*)d_in[15];
    const float* W2     = (const float*)d_in[16];
    const float* b2     = (const float*)d_in[17];

    float* ws  = (float*)d_ws;
    float* G0  = ws;                       // 2048*256
    float* G1  = G0 + (size_t)BT * NG;     // 2048*256
    float* H0  = G1 + (size_t)BT * NG;     // 2048*64
    float* H1  = H0 + (size_t)BT * E;      // 2048*64
    float* knr = H1 + (size_t)BT * E;      // 32*64
    float* kni = knr + (size_t)U * E;      // 32*64

    gemm_gates_kernel<<<512, 128, 0, stream>>>(x0, umask, W_ih0, b_ih0, b_hh0, G0, 300);
    gemm_gates_kernel<<<512, 128, 0, stream>>>(x1, umask, W_ih1, b_ih1, b_hh1, G1, 100);
    knorm_kernel<<<1, 32, 0, stream>>>(mkern, knr, kni);
    lstm_kernel<<<dim3(4, 2), 256, 0, stream>>>(G0, G1, W_hh0, W_hh1, H0, H1);
    tail_kernel<<<BT, 32, 0, stream>>>(H0, H1, smask, ptab, knr, kni,
                                       W1, b1, W2, b2, (float*)d_out);
}